// offset_Loss_45131516346379
// MI455X (gfx1250) — compile-verified
//
#include <hip/hip_runtime.h>

#define B_    8
#define W_    1024
#define H_    1024
#define L_    100
#define WH    (W_*H_)
#define NTOT  (B_*WH)
#define EPS_N 0.8388608f   /* 1e-7 * 8388608 */

typedef float v2f __attribute__((ext_vector_type(2)));
typedef float v8f __attribute__((ext_vector_type(8)));
typedef unsigned long long u64;

// ---------------- workspace layout (u32 units) ----------------
// [0,800)      cnt[b][l]            u32
// [800,2400)   packed (y<<32|x)     u64[800]  (sum_x<2^30, sum_y<2^30: no carry)
// [2400,3200)  mx bits[b][l]        u32 (float bit pattern, >=0)
// [3200,4000)  my bits[b][l]        u32
// [4000,4800)  xc[b][l]             f32
// [4800,5600)  yc[b][l]             f32
// [5600,5602)  fg_den, bg_den       f32
// [5632,9728)  block partials       f32 (512*8 = 4096)

__global__ void k_zero(unsigned* __restrict__ ws) {
  int i = blockIdx.x * blockDim.x + threadIdx.x;
  for (; i < 5 * B_ * L_; i += gridDim.x * blockDim.x) ws[i] = 0u;
}

__global__ void k_accum(const int* __restrict__ mask, unsigned* __restrict__ ws) {
  __shared__ unsigned s_cnt[L_];
  __shared__ u64      s_xy[L_];
  const int tid = threadIdx.x, b = blockIdx.y;
  for (int i = tid; i < L_; i += blockDim.x) { s_cnt[i] = 0u; s_xy[i] = 0ull; }
  __syncthreads();
  const int* mb = mask + (size_t)b * WH;
  const int pbase = blockIdx.x * 2048;
  const int4* mv = (const int4*)(mb + pbase);          // 16B-aligned b128 loads
#pragma unroll
  for (int g = 0; g < 2; ++g) {
    int j = g * 256 + tid;
    int4 s4 = mv[j];
    int p = pbase + j * 4;                             // 4 pixels, same row
    u64 ybase = ((u64)(unsigned)(p >> 10)) << 32;
    unsigned xb = (unsigned)(p & (H_ - 1));
    atomicAdd(&s_cnt[s4.x], 1u); atomicAdd(&s_xy[s4.x], ybase | (xb + 0u));
    atomicAdd(&s_cnt[s4.y], 1u); atomicAdd(&s_xy[s4.y], ybase | (xb + 1u));
    atomicAdd(&s_cnt[s4.z], 1u); atomicAdd(&s_xy[s4.z], ybase | (xb + 2u));
    atomicAdd(&s_cnt[s4.w], 1u); atomicAdd(&s_xy[s4.w], ybase | (xb + 3u));
  }
  __syncthreads();
  unsigned* g_cnt = ws;
  u64* g_xy = (u64*)(ws + 800);
  for (int i = tid; i < L_; i += blockDim.x) {
    if (s_cnt[i]) {
      atomicAdd(&g_cnt[b * L_ + i], s_cnt[i]);
      atomicAdd(&g_xy[b * L_ + i], s_xy[i]);           // global_atomic_add_u64
    }
  }
}

__global__ void k_centroid(unsigned* __restrict__ ws) {
  const unsigned* g_cnt = ws;
  const u64* g_xy = (const u64*)(ws + 800);
  float* xc   = (float*)(ws + 4000);
  float* yc   = (float*)(ws + 4800);
  float* dens = (float*)(ws + 5600);
  const int tid = threadIdx.x;
  for (int i = tid; i < B_ * L_; i += blockDim.x) {
    float sc = fmaxf((float)g_cnt[i], 1.0f);
    u64 pk = g_xy[i];
    xc[i] = (float)(unsigned)(pk & 0xffffffffu) / sc;
    yc[i] = (float)(unsigned)(pk >> 32) / sc;
  }
  if (tid == 0) {
    unsigned bgc = 0u;
    for (int b = 0; b < B_; ++b) bgc += g_cnt[b * L_]; // label-0 pixels
    float bg = (float)bgc, fg = (float)NTOT - bg;
    dens[0] = fg + EPS_N;
    dens[1] = bg + EPS_N;
  }
}

__global__ void k_vmax(const int* __restrict__ mask, unsigned* __restrict__ ws) {
  __shared__ float s_xc[L_], s_yc[L_];
  __shared__ unsigned s_mx[L_], s_my[L_];
  const int tid = threadIdx.x, b = blockIdx.y;
  const float* xc = (const float*)(ws + 4000);
  const float* yc = (const float*)(ws + 4800);
  for (int i = tid; i < L_; i += blockDim.x) {
    s_xc[i] = xc[b * L_ + i]; s_yc[i] = yc[b * L_ + i];
    s_mx[i] = 0u; s_my[i] = 0u;
  }
  __syncthreads();
  const int* mb = mask + (size_t)b * WH;
  const int pbase = blockIdx.x * 2048;
  const int4* mv = (const int4*)(mb + pbase);
#pragma unroll
  for (int g = 0; g < 2; ++g) {
    int j = g * 256 + tid;
    int4 s4 = mv[j];
    int p = pbase + j * 4;
    float y  = (float)(p >> 10);
    float xb = (float)(p & (H_ - 1));
    const int   ss[4] = { s4.x, s4.y, s4.z, s4.w };
#pragma unroll
    for (int k = 0; k < 4; ++k) {
      int s = ss[k];
      if (s != 0) {
        float vx = (xb + (float)k) - s_xc[s];
        float vy = y - s_yc[s];
        // only positive candidates matter (final mx = max(segmax, 0));
        // positive-float bit patterns order like u32 -> exact & deterministic
        if (vx > 0.f) atomicMax(&s_mx[s], __float_as_uint(vx));
        if (vy > 0.f) atomicMax(&s_my[s], __float_as_uint(vy));
      }
    }
  }
  __syncthreads();
  unsigned* g_mx = ws + 2400;
  unsigned* g_my = ws + 3200;
  for (int i = tid; i < L_; i += blockDim.x) {
    if (s_mx[i]) atomicMax(&g_mx[b * L_ + i], s_mx[i]);
    if (s_my[i]) atomicMax(&g_my[b * L_ + i], s_my[i]);
  }
}

__global__ void k_final(const float* __restrict__ pred, const int* __restrict__ mask,
                        const float* __restrict__ ig, float* __restrict__ out,
                        unsigned* __restrict__ ws) {
  __shared__ float s_xc[L_], s_yc[L_], s_ix[L_], s_iy[L_];
  __shared__ float s_wave[8];
  const int tid = threadIdx.x, b = blockIdx.y;
  const float* xc = (const float*)(ws + 4000);
  const float* yc = (const float*)(ws + 4800);
  const unsigned* g_mx = ws + 2400;
  const unsigned* g_my = ws + 3200;
  const float* dens = (const float*)(ws + 5600);
  for (int i = tid; i < L_; i += blockDim.x) {
    s_xc[i] = xc[b * L_ + i];
    s_yc[i] = yc[b * L_ + i];
    float mx = __uint_as_float(g_mx[b * L_ + i]);
    float my = __uint_as_float(g_my[b * L_ + i]);
    s_ix[i] = 1.0f / (mx == 0.f ? 1.f : mx);
    s_iy[i] = 1.0f / (my == 0.f ? 1.f : my);
  }
  __syncthreads();
  const float inv_fg = 1.0f / dens[0];
  const float inv_bg = 1.0f / dens[1];
  const float* p0  = pred + (size_t)b * 2 * WH;
  const float* p1  = p0 + WH;
  const float* igb = ig + (size_t)b * WH;
  const int*   mb  = mask + (size_t)b * WH;
  float* g0 = out + 1 + (size_t)b * 2 * WH;           // only 4B aligned (loss slot 0)
  float* g1 = g0 + WH;
  const int pbase = blockIdx.x * 2048;
  const float4* p0v = (const float4*)(p0 + pbase);    // 16B-aligned b128 streams
  const float4* p1v = (const float4*)(p1 + pbase);
  const float4* igv = (const float4*)(igb + pbase);
  const int4*   mv  = (const int4*)(mb + pbase);
  float acc = 0.f;
#pragma unroll
  for (int g = 0; g < 2; ++g) {
    int j = g * 256 + tid;
    if (g == 0) {                                     // gfx1250 global_prefetch_b8
      __builtin_prefetch(&p0v[j + 256], 0, 1);
      __builtin_prefetch(&p1v[j + 256], 0, 1);
    }
    int4   s4 = mv[j];
    float4 a0 = p0v[j];
    float4 a1 = p1v[j];
    float4 w4 = igv[j];
    int p = pbase + j * 4;
    float y  = (float)(p >> 10);
    float xb = (float)(p & (H_ - 1));
    const int   ss[4] = { s4.x, s4.y, s4.z, s4.w };
    const float pr0[4] = { a0.x, a0.y, a0.z, a0.w };
    const float pr1[4] = { a1.x, a1.y, a1.z, a1.w };
    const float wg[4]  = { w4.x, w4.y, w4.z, w4.w };
#pragma unroll
    for (int k = 0; k < 4; ++k) {
      int s = ss[k];
      float gx = 0.f, gy = 0.f;
      if (s != 0) {
        gx = ((xb + (float)k) - s_xc[s]) * s_ix[s];
        gy = (y - s_yc[s]) * s_iy[s];
      }
      g0[p + k] = gx;
      g1[p + k] = gy;
      float d0 = pr0[k] - gx, d1 = pr1[k] - gy;
      float q0 = fabsf(d0),   q1 = fabsf(d1);
      float l0 = (q0 < 1.f) ? 0.5f * d0 * d0 : q0 - 0.5f;
      float l1 = (q1 < 1.f) ? 0.5f * d1 * d1 : q1 - 0.5f;
      float wgt = wg[k] * ((s != 0) ? inv_fg : inv_bg);
      acc = fmaf(l0 + l1, wgt, acc);
    }
  }
  // ---- wave32 cross-lane reduction via the matrix engine ----
  // A(16x4): VGPR0 = lane partial (K=0 lanes 0-15, K=2 lanes 16-31), VGPR1 = 0.
  // B(4x16) = ones  =>  D[m][*] = acc(lane m) + acc(lane m+16).
  // Unconditional: EXEC all ones as WMMA requires.
  v2f a;  a[0] = acc; a[1] = 0.f;
  v2f bb; bb[0] = 1.f; bb[1] = 1.f;
  v8f c = {0.f, 0.f, 0.f, 0.f, 0.f, 0.f, 0.f, 0.f};
  v8f d = __builtin_amdgcn_wmma_f32_16x16x4_f32(false, a, false, bb,
                                                (short)0, c, false, false);
  float t = d[0] + d[1] + d[2] + d[3] + d[4] + d[5] + d[6] + d[7];
  float wave_total = __shfl(t, 0, 32) + __shfl(t, 16, 32);
  const int wid = tid >> 5;
  if ((tid & 31) == 0) s_wave[wid] = wave_total;
  __syncthreads();
  if (tid == 0) {
    float bsum = 0.f;
    for (int w = 0; w < 8; ++w) bsum += s_wave[w];    // fixed order -> deterministic
    float* partials = (float*)(ws + 5632);
    partials[blockIdx.y * gridDim.x + blockIdx.x] = bsum;
  }
}

__global__ void k_reduce(const unsigned* __restrict__ ws, float* __restrict__ out) {
  __shared__ float s[256];
  const float* partials = (const float*)(ws + 5632);
  const int tid = threadIdx.x;
  float a = 0.f;
  for (int i = tid; i < 4096; i += 256) a += partials[i];   // fixed order
  s[tid] = a;
  __syncthreads();
  for (int off = 128; off > 0; off >>= 1) {
    if (tid < off) s[tid] += s[tid + off];
    __syncthreads();
  }
  if (tid == 0) out[0] = s[0];
}

extern "C" void kernel_launch(void* const* d_in, const int* in_sizes, int n_in,
                              void* d_out, int out_size, void* d_ws, size_t ws_size,
                              hipStream_t stream) {
  const float* pred = (const float*)d_in[0];   // (B,2,W,H) f32
  const int*   mask = (const int*)d_in[1];     // (B,W,H)   i32
  const float* ig   = (const float*)d_in[2];   // (B,W,H)   f32
  float* out = (float*)d_out;                  // [loss, gt(B,2,W,H)]
  unsigned* ws = (unsigned*)d_ws;

  dim3 grid(WH / 2048, B_);                    // 512 x 8 blocks, 256 thr (8 waves)
  k_zero    <<<16,   256, 0, stream>>>(ws);
  k_accum   <<<grid, 256, 0, stream>>>(mask, ws);
  k_centroid<<<1,    256, 0, stream>>>(ws);
  k_vmax    <<<grid, 256, 0, stream>>>(mask, ws);
  k_final   <<<grid, 256, 0, stream>>>(pred, mask, ig, out, ws);
  k_reduce  <<<1,    256, 0, stream>>>(ws, out);
}